// GCN_65343632441566
// MI455X (gfx1250) — compile-verified
//
#include <hip/hip_runtime.h>

// ---------------------------------------------------------------------------
// 2-layer GCN for MI455X (gfx1250, wave32).
//   GEMMs:    V_WMMA_F32_16X16X4_F32 (fp32 in/out, matches reference precision)
//   Scatter:  float4 gathers + global_atomic_add_f32 (hardware fp atomics)
// Roofline: edge scatter dominates (~1.6 GB of mostly L2-resident traffic,
// ~70us floor at HBM rate, better from 192MB L2); GEMMs are ~2.5 GFLOP total,
// negligible, so fp32 WMMA (full reference precision) is the right choice.
//
// LDS layout for W: pair-interleaved, element (k,n) at
//   sW[(k/2)*PITCH + 2n + (k&1)]
// so each B fragment (rows k+koff, k+koff+1, one column) is a single aligned
// ds_load_b64 into an even VGPR pair -> no repacking movs before v_wmma.
// PITCH=160 floats: half-wave 1 reads pair-row p+1 => +32 bank rotation, so
// lanes 0-15 and 16-31 hit disjoint 32-bank sets (conflict-free wave32 b64).
// ---------------------------------------------------------------------------

typedef __attribute__((ext_vector_type(2))) float v2f;
typedef __attribute__((ext_vector_type(8))) float v8f;

#define IN_DIM  128
#define HID_DIM 64
#define OUT_DIM 64
#define W_PITCH 160   // floats per pair-row in LDS (128 data + 32 pad)

__global__ __launch_bounds__(256) void zero_f4(float4* __restrict__ p, int n4) {
  int i = blockIdx.x * blockDim.x + threadIdx.x;
  if (i < n4) p[i] = make_float4(0.f, 0.f, 0.f, 0.f);
}

// Y[N x 64] = X[N x K] @ W[K x 64].  One wave per 16-row tile of X; four
// 16x16 WMMA accumulators span all 64 output columns. K stepped by 4.
template <int K>
__global__ __launch_bounds__(256) void gemm_wmma(const float* __restrict__ X,
                                                 const float* __restrict__ W,
                                                 float* __restrict__ Y,
                                                 int nTiles) {
  __shared__ float sW[(K / 2) * W_PITCH];
  for (int i = threadIdx.x; i < K * 64; i += 256) {
    int r = i >> 6, c = i & 63;
    sW[(r >> 1) * W_PITCH + (c << 1) + (r & 1)] = W[i];
  }
  __syncthreads();

  const int wave = threadIdx.x >> 5;
  const int lane = threadIdx.x & 31;
  const int tile = blockIdx.x * 8 + wave;
  if (tile >= nTiles) return;           // wave-uniform: EXEC stays all-1s

  const int mrow = lane & 15;           // M (for A) / N (for B,D) position
  const int half = lane >> 4;           // 0 or 1
  const int koff = half << 1;           // half-wave K offset: 0 or 2

  const float* xrow = X + (size_t)(tile * 16 + mrow) * K;
  // LDS base for this lane's B fragments: pair-row (koff/2) = half, column 2*mrow
  const float* wbase = sW + (size_t)half * W_PITCH + (mrow << 1);

  v8f acc0 = {}, acc1 = {}, acc2 = {}, acc3 = {};

  for (int k = 0; k < K; k += 4) {
    // A fragment: rows of X, K=k+koff and k+koff+1 (even offset -> aligned b64)
    v2f a = *(const v2f*)(xrow + k + koff);
    // B fragments: single aligned ds_load_b64 each (pair-interleaved layout)
    const float* wk = wbase + (k >> 1) * W_PITCH;
    v2f b0 = *(const v2f*)(wk + 0);
    v2f b1 = *(const v2f*)(wk + 32);
    v2f b2 = *(const v2f*)(wk + 64);
    v2f b3 = *(const v2f*)(wk + 96);

    acc0 = __builtin_amdgcn_wmma_f32_16x16x4_f32(false, a, false, b0, (short)0, acc0, false, false);
    acc1 = __builtin_amdgcn_wmma_f32_16x16x4_f32(false, a, false, b1, (short)0, acc1, false, false);
    acc2 = __builtin_amdgcn_wmma_f32_16x16x4_f32(false, a, false, b2, (short)0, acc2, false, false);
    acc3 = __builtin_amdgcn_wmma_f32_16x16x4_f32(false, a, false, b3, (short)0, acc3, false, false);
  }

  // D layout: lane L holds column L%16 of rows (tile*16 + 8*(L/16) + v), v=0..7
  const int rbase = tile * 16 + (half << 3);
  float* y = Y + (size_t)rbase * 64 + mrow;
#pragma unroll
  for (int v = 0; v < 8; v++) {
    y[(size_t)v * 64 + 0]  = acc0[v];
    y[(size_t)v * 64 + 16] = acc1[v];
    y[(size_t)v * 64 + 32] = acc2[v];
    y[(size_t)v * 64 + 48] = acc3[v];
  }
}

// agg[dst[e]] += H[src[e]] * ew[e]; 16 threads per edge, float4 per thread.
__global__ __launch_bounds__(256) void scatter_edges(const float* __restrict__ H,
                                                     const int* __restrict__ src,
                                                     const int* __restrict__ dst,
                                                     const float* __restrict__ ew,
                                                     float* __restrict__ agg,
                                                     int nEdges) {
  long long t = (long long)blockIdx.x * 256 + threadIdx.x;
  int e  = (int)(t >> 4);
  int f4 = (int)(t & 15);
  if (e >= nEdges) return;
  int s = src[e], d = dst[e];
  float w = ew[e];
  float4 v = reinterpret_cast<const float4*>(H + (size_t)s * 64)[f4];
  float* p = agg + (size_t)d * 64 + (f4 << 2);
  unsafeAtomicAdd(p + 0, v.x * w);
  unsafeAtomicAdd(p + 1, v.y * w);
  unsafeAtomicAdd(p + 2, v.z * w);
  unsafeAtomicAdd(p + 3, v.w * w);
}

// out = relu(agg + b), float4-vectorized over [N,64]
__global__ __launch_bounds__(256) void relu_bias_k(const float4* __restrict__ agg,
                                                   const float* __restrict__ b,
                                                   float4* __restrict__ out, int n4) {
  int i = blockIdx.x * blockDim.x + threadIdx.x;
  if (i >= n4) return;
  float4 bv = reinterpret_cast<const float4*>(b)[i & 15];
  float4 v = agg[i];
  v.x = fmaxf(v.x + bv.x, 0.f);
  v.y = fmaxf(v.y + bv.y, 0.f);
  v.z = fmaxf(v.z + bv.z, 0.f);
  v.w = fmaxf(v.w + bv.w, 0.f);
  out[i] = v;
}

// out += b (broadcast over rows), in-place final bias
__global__ __launch_bounds__(256) void bias_add_k(float4* __restrict__ out,
                                                  const float* __restrict__ b, int n4) {
  int i = blockIdx.x * blockDim.x + threadIdx.x;
  if (i >= n4) return;
  float4 bv = reinterpret_cast<const float4*>(b)[i & 15];
  float4 v = out[i];
  v.x += bv.x; v.y += bv.y; v.z += bv.z; v.w += bv.w;
  out[i] = v;
}

extern "C" void kernel_launch(void* const* d_in, const int* in_sizes, int n_in,
                              void* d_out, int out_size, void* d_ws, size_t ws_size,
                              hipStream_t stream) {
  const float* x  = (const float*)d_in[0];
  const int*   ei = (const int*)d_in[1];
  const float* ew = (const float*)d_in[2];
  const float* w1 = (const float*)d_in[3];
  const float* b1 = (const float*)d_in[4];
  const float* w2 = (const float*)d_in[5];
  const float* b2 = (const float*)d_in[6];

  const int N = in_sizes[0] / IN_DIM;   // 100000
  const int E = in_sizes[2];            // 1600000
  const int* src = ei;
  const int* dst = ei + E;
  float* out = (float*)d_out;

  float* A = (float*)d_ws;              // [N,64] h1, then relu(h1)
  float* B = A + (size_t)N * 64;        // [N,64] agg1, then h2

  const int n4 = N * 16;                // #float4 in an [N,64] buffer
  const int zb = (n4 + 255) / 256;
  const int nTiles = (N + 15) / 16;
  const int gemmBlocks = (nTiles + 7) / 8;
  const long long st = (long long)E * 16;
  const int sb = (int)((st + 255) / 256);

  // Layer 1
  zero_f4<<<zb, 256, 0, stream>>>((float4*)B, n4);
  gemm_wmma<IN_DIM><<<gemmBlocks, 256, 0, stream>>>(x, w1, A, nTiles);
  scatter_edges<<<sb, 256, 0, stream>>>(A, src, dst, ew, B, E);
  relu_bias_k<<<zb, 256, 0, stream>>>((const float4*)B, b1, (float4*)A, n4);

  // Layer 2
  zero_f4<<<zb, 256, 0, stream>>>((float4*)out, n4);
  gemm_wmma<HID_DIM><<<gemmBlocks, 256, 0, stream>>>(A, w2, B, nTiles);
  scatter_edges<<<sb, 256, 0, stream>>>(B, src, dst, ew, out, E);
  bias_add_k<<<zb, 256, 0, stream>>>((float4*)out, b2, n4);
}